// SelfAttention_27994596835548
// MI455X (gfx1250) — compile-verified
//
#include <hip/hip_runtime.h>
#include <hip/hip_bf16.h>
#include <stdint.h>

typedef __attribute__((ext_vector_type(2))) float v2f;
typedef __attribute__((ext_vector_type(8))) float v8f;

#define BATCH 4
#define CCH   256
#define CQ    32
#define NPIX  4096

static __device__ __forceinline__ v8f wmma4(v2f a, v2f b, v8f c) {
    return __builtin_amdgcn_wmma_f32_16x16x4_f32(false, a, false, b,
                                                 (short)0, c, false, false);
}

// ---- CDNA5 async global->LDS copy (ASYNCcnt) with synchronous fallback ----
#if defined(__has_builtin)
#if __has_builtin(__builtin_amdgcn_global_load_async_to_lds_b128) && \
    __has_builtin(__builtin_amdgcn_s_wait_asynccnt)
#define USE_ASYNC_LDS 1
#endif
#endif

#ifdef USE_ASYNC_LDS
typedef int v4i_vs __attribute__((vector_size(4 * sizeof(int))));
typedef __attribute__((address_space(1))) v4i_vs* g128p;   // global b128 ptr
typedef __attribute__((address_space(3))) v4i_vs* l128p;   // LDS    b128 ptr
#endif

static __device__ __forceinline__ void copy16_to_lds(const float* g, float* l) {
#ifdef USE_ASYNC_LDS
    // LDS flat address: low 32 bits are the LDS byte offset (ISA 10.2).
    __builtin_amdgcn_global_load_async_to_lds_b128(
        (g128p)(uintptr_t)g,
        (l128p)(uintptr_t)(uint32_t)(uintptr_t)l,
        0, 0);
#else
    *(float4*)l = *(const float4*)g;
#endif
}

static __device__ __forceinline__ void wait_async(int n) {
#ifdef USE_ASYNC_LDS
    if (n == 0) __builtin_amdgcn_s_wait_asynccnt(0);
    else        __builtin_amdgcn_s_wait_asynccnt(5);
#endif
    (void)n;
}

// ---------------------------------------------------------------------------
// Kernel 1: Out[b,m,n] = sum_k W[m,k] * X[b,k,n] + bias[m]
// 16(M) x 64(N) tile per wave; B loads lane-coalesced (X[k*N + n]).
// ---------------------------------------------------------------------------
template <int M>
__global__ __launch_bounds__(256) void k_qkv_gemm(
    const float* __restrict__ W, const float* __restrict__ bias,
    const float* __restrict__ X, float* __restrict__ Out)
{
    constexpr int K  = CCH;
    constexpr int N  = NPIX;
    constexpr int mt = M / 16;
    constexpr int nt = N / 64;
    constexpr int total = BATCH * mt * nt;

    const int lane = threadIdx.x & 31;
    const int half = lane >> 4;
    const int lq   = lane & 15;
    const int gw = blockIdx.x * (blockDim.x >> 5) + (threadIdx.x >> 5);
    const int nw = gridDim.x * (blockDim.x >> 5);

    for (int t = gw; t < total; t += nw) {
        const int b  = t / (mt * nt);
        const int r  = t - b * (mt * nt);
        const int mi = r / nt;
        const int ni = r - mi * nt;
        const float* Xb = X   + (size_t)b * K * N;
        float*       Ob = Out + (size_t)b * M * N;
        const int row0 = mi << 4, col0 = ni << 6;
        const int m = row0 + lq;
        const int n = col0 + lq;

        const float* Ap = W  + (size_t)m * K;
        const float* Bp = Xb + n;

        v8f acc0 = {}, acc1 = {}, acc2 = {}, acc3 = {};
        for (int kk = 0; kk < K; kk += 4) {
            const int kb = kk + half * 2;
            const v2f a = *(const v2f*)(Ap + kb);
            const float* b0 = Bp + (size_t)kb * N;
            v2f f0; f0.x = b0[ 0]; f0.y = b0[N +  0];
            v2f f1; f1.x = b0[16]; f1.y = b0[N + 16];
            v2f f2; f2.x = b0[32]; f2.y = b0[N + 32];
            v2f f3; f3.x = b0[48]; f3.y = b0[N + 48];
            acc0 = wmma4(a, f0, acc0);
            acc1 = wmma4(a, f1, acc1);
            acc2 = wmma4(a, f2, acc2);
            acc3 = wmma4(a, f3, acc3);
        }
#pragma unroll
        for (int j = 0; j < 8; ++j) {
            const int row = row0 + j + half * 8;
            const float bs = bias[row];
            float* o = Ob + (size_t)row * N + n;
            o[ 0] = acc0[j] + bs;
            o[16] = acc1[j] + bs;
            o[32] = acc2[j] + bs;
            o[48] = acc3[j] + bs;
        }
    }
}

// ---------------------------------------------------------------------------
// Kernel 2: ST[b,j,i] = sum_c K[b,c,j] * Q[b,c,i]  (scores, pre-transposed
// into attention_map layout).  All fragment loads lane-coalesced.
// ---------------------------------------------------------------------------
__global__ __launch_bounds__(256) void k_scores(
    const float* __restrict__ Q, const float* __restrict__ Km,
    float* __restrict__ ST)
{
    constexpr int nt_j = NPIX / 16;
    constexpr int nt_i = NPIX / 64;
    constexpr int total = BATCH * nt_j * nt_i;

    const int lane = threadIdx.x & 31;
    const int half = lane >> 4;
    const int lq   = lane & 15;
    const int gw = blockIdx.x * (blockDim.x >> 5) + (threadIdx.x >> 5);
    const int nw = gridDim.x * (blockDim.x >> 5);

    for (int t = gw; t < total; t += nw) {
        const int b  = t / (nt_j * nt_i);
        const int r  = t - b * (nt_j * nt_i);
        const int ji = r / nt_i;
        const int ii = r - ji * nt_i;
        const float* Qb = Q  + (size_t)b * CQ * NPIX;
        const float* Kb = Km + (size_t)b * CQ * NPIX;
        float*       Sb = ST + (size_t)b * NPIX * NPIX;
        const int j0 = ji << 4, i0 = ii << 6;
        const int jrow = j0 + lq;
        const int icol = i0 + lq;

        v8f acc0 = {}, acc1 = {}, acc2 = {}, acc3 = {};
#pragma unroll
        for (int kk = 0; kk < CQ; kk += 4) {
            const int kb = kk + half * 2;
            const float* ka = Kb + (size_t)kb * NPIX;
            v2f a;  a.x  = ka[jrow];        a.y  = ka[NPIX + jrow];
            const float* qb = Qb + (size_t)kb * NPIX + icol;
            v2f f0; f0.x = qb[ 0]; f0.y = qb[NPIX +  0];
            v2f f1; f1.x = qb[16]; f1.y = qb[NPIX + 16];
            v2f f2; f2.x = qb[32]; f2.y = qb[NPIX + 32];
            v2f f3; f3.x = qb[48]; f3.y = qb[NPIX + 48];
            acc0 = wmma4(a, f0, acc0);
            acc1 = wmma4(a, f1, acc1);
            acc2 = wmma4(a, f2, acc2);
            acc3 = wmma4(a, f3, acc3);
        }
#pragma unroll
        for (int j = 0; j < 8; ++j) {
            const int row = j0 + j + half * 8;
            float* o = Sb + (size_t)row * NPIX + icol;
            o[ 0] = acc0[j];
            o[16] = acc1[j];
            o[32] = acc2[j];
            o[48] = acc3[j];
        }
    }
}

// ---------------------------------------------------------------------------
// Kernel 3: in-place softmax over contiguous rows of 4096.
// ---------------------------------------------------------------------------
__global__ __launch_bounds__(256) void k_softmax(float* __restrict__ ST)
{
    __shared__ float red[256];
    const int tid = threadIdx.x;
    float4* p4 = (float4*)(ST + (size_t)blockIdx.x * NPIX);

    float4 v[4];
    float lmax = -3.402823466e38f;
#pragma unroll
    for (int i = 0; i < 4; ++i) {
        v[i] = p4[i * 256 + tid];
        lmax = fmaxf(lmax, fmaxf(fmaxf(v[i].x, v[i].y), fmaxf(v[i].z, v[i].w)));
    }
    red[tid] = lmax;
    __syncthreads();
    for (int s = 128; s > 0; s >>= 1) {
        if (tid < s) red[tid] = fmaxf(red[tid], red[tid + s]);
        __syncthreads();
    }
    const float rmax = red[0];
    __syncthreads();

    float lsum = 0.f;
#pragma unroll
    for (int i = 0; i < 4; ++i) {
        v[i].x = __expf(v[i].x - rmax);
        v[i].y = __expf(v[i].y - rmax);
        v[i].z = __expf(v[i].z - rmax);
        v[i].w = __expf(v[i].w - rmax);
        lsum += v[i].x + v[i].y + v[i].z + v[i].w;
    }
    red[tid] = lsum;
    __syncthreads();
    for (int s = 128; s > 0; s >>= 1) {
        if (tid < s) red[tid] += red[tid + s];
        __syncthreads();
    }
    const float inv = 1.0f / red[0];
#pragma unroll
    for (int i = 0; i < 4; ++i) {
        v[i].x *= inv; v[i].y *= inv; v[i].z *= inv; v[i].w *= inv;
        p4[i * 256 + tid] = v[i];
    }
}

// ---------------------------------------------------------------------------
// Kernel 4 (dominant GEMM, double-buffered async LDS staging):
//   o[b,c,n] = sum_m V[b,c,m] * AM[b,n,m];  out = x + gamma*o.
// 512 threads (16 waves) compute 256(c) x 64(n); AM fetched from global
// exactly once.  k-slice KB=32, two LDS buffers (90 KB total), async DMA of
// slice i+1 overlaps the 32-WMMA burst on slice i.
// ---------------------------------------------------------------------------
#define KB  32
#define LDA 36    // padded row stride: 16B-aligned float4, conflict-free banks

__global__ __launch_bounds__(512) void k_out_gemm(
    const float* __restrict__ V, const float* __restrict__ AM,
    const float* __restrict__ X, const float* __restrict__ gamma,
    float* __restrict__ Out)
{
    __shared__ float As[2][CCH * LDA];   // 2 x 36864 B
    __shared__ float Bs[2][64  * LDA];   // 2 x  9216 B

    const int tid  = threadIdx.x;
    const int wid  = tid >> 5;
    const int lane = tid & 31;
    const int half = lane >> 4;
    const int lq   = lane & 15;

    constexpr int nb = NPIX / 64;
    const int b   = blockIdx.x / nb;
    const int niB = blockIdx.x - b * nb;
    const int n0  = niB << 6;

    const float* Vb = V  + (size_t)b * CCH * NPIX;
    const float* Ab = AM + (size_t)b * NPIX * NPIX;
    const size_t xoff = (size_t)b * CCH * NPIX;

    // staging geometry: A = 2048 float4 units (4/thread), B = 512 (1/thread)
    const int rowA[4] = { (tid + 0 * 512) >> 3, (tid + 1 * 512) >> 3,
                          (tid + 2 * 512) >> 3, (tid + 3 * 512) >> 3 };
    const int kqA     = (tid & 7) << 2;
    const int rowB    = tid >> 3;

    // prologue: stage slice 0 into buffer 0 (5 async ops per wave)
    {
#pragma unroll
        for (int u = 0; u < 4; ++u)
            copy16_to_lds(Vb + (size_t)rowA[u] * NPIX + kqA,
                          &As[0][rowA[u] * LDA + kqA]);
        copy16_to_lds(Ab + (size_t)(n0 + rowB) * NPIX + kqA,
                      &Bs[0][rowB * LDA + kqA]);
    }

    v8f acc0 = {}, acc1 = {}, acc2 = {}, acc3 = {};

    for (int kk = 0; kk < NPIX; kk += KB) {
        const int cur = (kk / KB) & 1;
        const int nxt = cur ^ 1;

        __syncthreads();                      // buffer `nxt` free for restage
        if (kk + KB < NPIX) {
#pragma unroll
            for (int u = 0; u < 4; ++u)
                copy16_to_lds(Vb + (size_t)rowA[u] * NPIX + (kk + KB) + kqA,
                              &As[nxt][rowA[u] * LDA + kqA]);
            copy16_to_lds(Ab + (size_t)(n0 + rowB) * NPIX + (kk + KB) + kqA,
                          &Bs[nxt][rowB * LDA + kqA]);
            wait_async(5);                    // retire `cur` ops, keep prefetch
        } else {
            wait_async(0);
        }
        __syncthreads();                      // `cur` visible to all waves

        const float* Ar = &As[cur][(wid * 16 + lq) * LDA];
        const float* Br = &Bs[cur][lq * LDA];
#pragma unroll
        for (int k4 = 0; k4 < KB; k4 += 4) {
            const int kb = k4 + half * 2;
            const v2f a  = *(const v2f*)(Ar + kb);
            const v2f f0 = *(const v2f*)(Br + kb);
            const v2f f1 = *(const v2f*)(Br + 16 * LDA + kb);
            const v2f f2 = *(const v2f*)(Br + 32 * LDA + kb);
            const v2f f3 = *(const v2f*)(Br + 48 * LDA + kb);
            acc0 = wmma4(a, f0, acc0);
            acc1 = wmma4(a, f1, acc1);
            acc2 = wmma4(a, f2, acc2);
            acc3 = wmma4(a, f3, acc3);
        }
    }

    const float g = gamma[0];
    const int c0 = wid << 4;
    const int n  = n0 + lq;
#pragma unroll
    for (int j = 0; j < 8; ++j) {
        const int row = c0 + j + half * 8;
        const size_t idx = xoff + (size_t)row * NPIX + n;
        Out[idx +  0] = X[idx +  0] + g * acc0[j];
        Out[idx + 16] = X[idx + 16] + g * acc1[j];
        Out[idx + 32] = X[idx + 32] + g * acc2[j];
        Out[idx + 48] = X[idx + 48] + g * acc3[j];
    }
}

// ---------------------------------------------------------------------------
extern "C" void kernel_launch(void* const* d_in, const int* in_sizes, int n_in,
                              void* d_out, int out_size, void* d_ws, size_t ws_size,
                              hipStream_t stream)
{
    const float* x     = (const float*)d_in[0];
    const float* Wq    = (const float*)d_in[1];
    const float* bq    = (const float*)d_in[2];
    const float* Wk    = (const float*)d_in[3];
    const float* bk    = (const float*)d_in[4];
    const float* Wv    = (const float*)d_in[5];
    const float* bv    = (const float*)d_in[6];
    const float* gamma = (const float*)d_in[7];

    float* out  = (float*)d_out;
    float* attn = out + (size_t)BATCH * CCH * NPIX;   // attention_map region

    float* q = (float*)d_ws;                          // [B, CQ, N]
    float* k = q + (size_t)BATCH * CQ * NPIX;         // [B, CQ, N]
    float* v = k + (size_t)BATCH * CQ * NPIX;         // [B, C,  N]

    k_qkv_gemm<CQ ><<<  512 / 8, 256, 0, stream>>>(Wq, bq, x, q);
    k_qkv_gemm<CQ ><<<  512 / 8, 256, 0, stream>>>(Wk, bk, x, k);
    k_qkv_gemm<CCH><<< 4096 / 8, 256, 0, stream>>>(Wv, bv, x, v);

    k_scores<<<65536 / 8, 256, 0, stream>>>(q, k, attn);

    k_softmax<<<BATCH * NPIX, 256, 0, stream>>>(attn);

    k_out_gemm<<<BATCH * (NPIX / 64), 512, 0, stream>>>(v, attn, x, gamma, out);
}